// PostProcessing_3590592659708
// MI455X (gfx1250) — compile-verified
//
#include <hip/hip_runtime.h>

// ---------------------------------------------------------------------------
// Panoptic post-processing for MI455X (gfx1250, wave32, WMMA + TDM).
// B=2, C=34, H=512, W=1024, K=200 centers, 7x7 NMS, thresh 0.1.
// ---------------------------------------------------------------------------

#define BN 2
#define CC 34
#define HH 512
#define WW 1024
#define HWP (HH * WW)
#define KK 200
#define KPAD 208              // 13 tiles of 16 centers
#define KTILES (KPAD / 16)
#define KSZ 7
#define NMS_THRESH 0.1f
#define CAP 65536             // max NMS survivors kept per batch
#define NBINS ((KK + 1) * CC) // (inst, class) histogram bins
#define CTE (3 * KPAD)        // center-table floats per batch: cx | cy | cnorm

// d_out layout (float elements), reference tuple order:
// inst_class, inst_prob, seg_prob, n_pix, valid_inst, segs, closest
#define O_CLASS 0
#define O_PROB  (BN * KK)
#define O_SPROB (2 * BN * KK)
#define O_NPIX  (3 * BN * KK)
#define O_VALID (4 * BN * KK)
#define O_SEG   (5 * BN * KK)
#define O_CLOSE (5 * BN * KK + BN * HWP)

typedef float v2f __attribute__((ext_vector_type(2)));
typedef float v8f __attribute__((ext_vector_type(8)));
typedef unsigned int v4u __attribute__((ext_vector_type(4)));
typedef int v4i __attribute__((ext_vector_type(4)));
typedef int v8i __attribute__((ext_vector_type(8)));

// ---------------------------------------------------------------------------
// 0) zero the atomic scratch regions (ws is poisoned 0xAA by the harness)
// ---------------------------------------------------------------------------
__global__ void k_init(int* __restrict__ candCnt, unsigned* __restrict__ counts,
                       float* __restrict__ sums) {
  int i = blockIdx.x * blockDim.x + threadIdx.x;
  if (i < BN) candCnt[i] = 0;
  if (i < BN * NBINS) counts[i] = 0u;
  if (i < BN * KK) sums[i] = 0.f;
}

// ---------------------------------------------------------------------------
// 1) fused softmax denom + argmax over C=34 channels (bandwidth-bound pass)
// ---------------------------------------------------------------------------
__global__ void __launch_bounds__(256)
k_softmax(const float* __restrict__ logits, float* __restrict__ out,
          float* __restrict__ maxlog, float* __restrict__ invden,
          int* __restrict__ segI) {
  int pix = blockIdx.x * blockDim.x + threadIdx.x;
  int b = blockIdx.y;
  if (pix >= HWP) return;
  const float* base = logits + (size_t)b * CC * HWP + pix;
  __builtin_prefetch(base + (size_t)(CC - 1) * HWP, 0, 0); // global_prefetch_b8
  float lg[CC];
#pragma unroll
  for (int c = 0; c < CC; ++c) lg[c] = base[(size_t)c * HWP];
  float m = lg[0];
  int am = 0;
#pragma unroll
  for (int c = 1; c < CC; ++c)
    if (lg[c] > m) { m = lg[c]; am = c; }   // first-max == jnp.argmax
  float s = 0.f;
#pragma unroll
  for (int c = 0; c < CC; ++c) s += expf(lg[c] - m);
  size_t p = (size_t)b * HWP + pix;
  maxlog[p] = m;
  invden[p] = 1.f / s;
  segI[p] = am;
  out[O_SEG + p] = (float)am;
}

// ---------------------------------------------------------------------------
// 2) 7x7 NMS max-pool via LDS tiles (16x16 tile + halo 3)
// ---------------------------------------------------------------------------
__global__ void __launch_bounds__(256)
k_nms(const float* __restrict__ cm, float* __restrict__ nms) {
  const int TS = 16, HALO = 3, LW = TS + 2 * HALO; // 22
  __shared__ float tile[LW * LW];
  int b = blockIdx.z;
  int tx = threadIdx.x, ty = threadIdx.y;
  int gx0 = blockIdx.x * TS - HALO, gy0 = blockIdx.y * TS - HALO;
  const float* src = cm + (size_t)b * HWP;
  int tid = ty * TS + tx;
  for (int i = tid; i < LW * LW; i += TS * TS) {
    int ly = i / LW, lx = i % LW;
    int gx = gx0 + lx, gy = gy0 + ly;
    float v = -__builtin_inff();
    if (gx >= 0 && gx < WW && gy >= 0 && gy < HH) v = src[gy * WW + gx];
    tile[i] = v;
  }
  __syncthreads();
  float center = tile[(ty + HALO) * LW + (tx + HALO)];
  float mx = -__builtin_inff();
#pragma unroll
  for (int dy = 0; dy < KSZ; ++dy)
#pragma unroll
    for (int dx = 0; dx < KSZ; ++dx)
      mx = fmaxf(mx, tile[(ty + dy) * LW + (tx + dx)]);
  int gx = blockIdx.x * TS + tx, gy = blockIdx.y * TS + ty;
  float outv = (mx > NMS_THRESH && center == mx) ? center : 0.f;
  nms[(size_t)b * HWP + gy * WW + gx] = outv;
}

// ---------------------------------------------------------------------------
// 3) compact NMS survivors (~H*W/49 expected) into candidate lists
// ---------------------------------------------------------------------------
__global__ void k_compact(const float* __restrict__ nms, int* __restrict__ cnt,
                          float* __restrict__ cv, int* __restrict__ ci) {
  int pix = blockIdx.x * blockDim.x + threadIdx.x;
  int b = blockIdx.y;
  if (pix >= HWP) return;
  float v = nms[(size_t)b * HWP + pix];
  if (v > 0.f) {
    int slot = atomicAdd(&cnt[b], 1);
    if (slot < CAP) {
      cv[(size_t)b * CAP + slot] = v;
      ci[(size_t)b * CAP + slot] = pix;
    }
  }
}

// ---------------------------------------------------------------------------
// 4) iterative top-K=200 selection, one workgroup per batch.
//    Comparator (value desc, pixel-index asc) is order-independent, so the
//    result is deterministic regardless of atomic compaction order.
//    ctab layout per batch: [0..KPAD) = cx, [KPAD..2K) = cy, [2K..3K) = |c|^2
// ---------------------------------------------------------------------------
__global__ void __launch_bounds__(256)
k_topk(const int* __restrict__ cnt, float* __restrict__ cv, const int* __restrict__ ci,
       float* __restrict__ probs, int* __restrict__ validk,
       float* __restrict__ ctab, float* __restrict__ out) {
  int b = blockIdx.x;
  int t = threadIdx.x;
  int n = cnt[b]; if (n > CAP) n = CAP;
  float* v = cv + (size_t)b * CAP;
  const int* id = ci + (size_t)b * CAP;
  float* ct = ctab + (size_t)b * CTE;
  __shared__ float sv[256];
  __shared__ int si[256];
  __shared__ int ss[256];
  for (int k = 0; k < KK; ++k) {
    float bv = 0.f; int bi = 0x7fffffff; int bs = -1;
    for (int i = t; i < n; i += 256) {
      float val = v[i];
      int pid = id[i];
      if (val > bv || (val == bv && val > 0.f && pid < bi)) { bv = val; bi = pid; bs = i; }
    }
    sv[t] = bv; si[t] = bi; ss[t] = bs;
    __syncthreads();
    for (int off = 128; off; off >>= 1) {
      if (t < off) {
        float ov = sv[t + off]; int oi = si[t + off];
        if (ov > sv[t] || (ov == sv[t] && ov > 0.f && oi < si[t])) {
          sv[t] = ov; si[t] = oi; ss[t] = ss[t + off];
        }
      }
      __syncthreads();
    }
    if (t == 0) {
      float pv = sv[0]; int pi = si[0]; int ps = ss[0];
      int ok = (ps >= 0);
      float xs = 0.f, ys = 0.f;
      if (ok) { v[ps] = -1.f; xs = (float)(pi % WW); ys = (float)(pi / WW); }
      probs[b * KK + k]        = ok ? pv : 0.f;
      validk[b * KK + k]       = ok;
      out[O_PROB + b * KK + k] = ok ? pv : 0.f;
      ct[k]            = ok ? xs : 0.f;
      ct[KPAD + k]     = ok ? ys : 0.f;
      ct[2 * KPAD + k] = ok ? (xs * xs + ys * ys) : __builtin_inff();
    }
    __threadfence_block();
    __syncthreads();
  }
  if (t < KPAD - KK) { // pad 200..207 as unreachable centers
    int k = KK + t;
    ct[k] = 0.f;
    ct[KPAD + k] = 0.f;
    ct[2 * KPAD + k] = __builtin_inff();
  }
}

// ---------------------------------------------------------------------------
// 5) WMMA closest-center kernel.
//    score(k) = |c_k|^2 - 2*(px*cx_k + py*cy_k)   (|p|^2 cancels in argmin)
//    Cross term via V_WMMA_F32_16X16X4_F32: 16 pixels x 16 centers per op.
//    A (16x4, documented): lanes 0-15 VGPR0=px(K=0), VGPR1=py(K=1);
//                          lanes 16-31 hold K=2,3 -> set to 0.
//    B (4x16): half-wave row order is not documented, so cy is placed in BOTH
//    candidate K=1 positions (VGPR0 lanes 16-31 and VGPR1 lanes 0-15). Since
//    A's K=2,3 slots are zero, either ordering yields px*cx + py*cy.
//    The selection is done BRANCHLESSLY: the x-operand picks cx vs cy purely
//    by per-lane LDS address (hoisted), the y-operand is masked by a hoisted
//    0/1 multiplier -- no EXEC manipulation inside the WMMA loop.
//    D (16x16 f32, documented): VGPR v = row m=v (lanes 0-15) / m=v+8 (16-31).
//    Center tables are DMA'd into LDS with the Tensor Data Mover.
// ---------------------------------------------------------------------------
__global__ void __launch_bounds__(256)
k_closest(const float* __restrict__ reg, const float* __restrict__ ctab,
          int* __restrict__ closestI, float* __restrict__ out) {
  __shared__ float sct[CTE];
  int b = blockIdx.y;
  int t = threadIdx.x;

#if defined(__gfx1250__) && __has_builtin(__builtin_amdgcn_tensor_load_to_lds)
  if (t < 32) { // one wave issues the TDM transfer (EXEC ignored by TDM)
    // D# group0: count=1, lds_addr, 57-bit global addr, type=2 (bits 127:126)
    unsigned ldsOff = (unsigned)(size_t)&sct[0]; // low 32b of generic = LDS offset
    unsigned long long ga =
        (unsigned long long)(size_t)(ctab + (size_t)b * CTE);
    v4u g0 = { 1u, ldsOff, (unsigned)ga,
               (unsigned)((ga >> 32) & 0x1FFFFFFull) | 0x80000000u };
    // D# group1: data_size=4B (code 2), tensor_dim0=tile_dim0=CTE (1-D row),
    // tensor_dim1=1, tile_dim1=1, tensor_dim0_stride=CTE, no pad/barrier.
    v8i g1 = { (int)(2u << 16),
               (int)(CTE << 16),     // tensor_dim0 lo16 << 16
               (int)(1 << 16),       // tensor_dim1 lo16 << 16
               (int)(CTE << 16),     // tile_dim0 << 16
               1,                    // tile_dim1
               (int)CTE,             // tensor_dim0_stride lo32
               0, 0 };
    v4i g2 = {0, 0, 0, 0}, g3 = {0, 0, 0, 0};
#if __clang_major__ >= 23
    v8i g4 = {0, 0, 0, 0, 0, 0, 0, 0};
    __builtin_amdgcn_tensor_load_to_lds(g0, g1, g2, g3, g4, 0);
#else
    __builtin_amdgcn_tensor_load_to_lds(g0, g1, g2, g3, 0);
#endif
    __builtin_amdgcn_s_wait_tensorcnt(0); // s_wait_tensorcnt 0x0
  }
  __syncthreads();
#else
  for (int i = t; i < CTE; i += 256) sct[i] = ctab[(size_t)b * CTE + i];
  __syncthreads();
#endif

  int wave = t >> 5, lane = t & 31;
  int half = lane >> 4, l16 = lane & 15;
  int pixBase = (blockIdx.x * 8 + wave) * 16;
  const float* reg0 = reg + (size_t)b * 2 * HWP;
  const float* reg1 = reg0 + HWP;

  float px = 0.f, py = 0.f;
  if (half == 0) {
    int pix = pixBase + l16;
    int w = pix & (WW - 1);
    int h = pix >> 10;
    px = (float)(w + 1) - reg0[pix];
    py = (float)(h + 1) - reg1[pix];
  }
  v2f A; A.x = px; A.y = py;

  // Hoisted, loop-invariant B-operand addressing (no branches in the loop):
  // Bm.x source: cx (half 0) or cy (half 1) selected by LDS offset.
  // Bm.y source: cy, zeroed for half 1 via multiplier.
  const int offX = (half ? KPAD : 0) + l16;
  const int offY = KPAD + l16;
  const int offN = 2 * KPAD + l16;
  const float ymul = half ? 0.f : 1.f;

  float bestV[8];
  int bestK[8];
#pragma unroll
  for (int v = 0; v < 8; ++v) { bestV[v] = __builtin_inff(); bestK[v] = 0; }

#pragma unroll
  for (int tt = 0; tt < KTILES; ++tt) {
    v2f Bm;
    Bm.x = sct[tt * 16 + offX];
    Bm.y = sct[tt * 16 + offY] * ymul;
    v8f Cz = {0.f, 0.f, 0.f, 0.f, 0.f, 0.f, 0.f, 0.f};
    v8f D = __builtin_amdgcn_wmma_f32_16x16x4_f32(
        false, A, false, Bm, (short)0, Cz, false, false);
    float cn = sct[tt * 16 + offN];
    int idx = tt * 16 + l16;
#pragma unroll
    for (int v = 0; v < 8; ++v) {
      float dist = cn - 2.f * D[v]; // row m = v + 8*half, col n = l16
      if (dist < bestV[v]) { bestV[v] = dist; bestK[v] = idx; }
    }
  }

  // argmin across the 16 columns (xor masks <=8 stay within each half-wave)
#pragma unroll
  for (int v = 0; v < 8; ++v) {
    float bv = bestV[v]; int bk = bestK[v];
#pragma unroll
    for (int off = 1; off < 16; off <<= 1) {
      float ov = __shfl_xor(bv, off, 32);
      int ok = __shfl_xor(bk, off, 32);
      if (ov < bv || (ov == bv && ok < bk)) { bv = ov; bk = ok; }
    }
    bestV[v] = bv; bestK[v] = bk;
  }

  if (l16 == 0) {
    int mbase = pixBase + half * 8;
#pragma unroll
    for (int v = 0; v < 8; ++v) {
      size_t p = (size_t)b * HWP + mbase + v;
      int cl = bestK[v] + 1; // ids 1..K
      closestI[p] = cl;
      out[O_CLOSE + p] = (float)cl;
    }
  }
}

// ---------------------------------------------------------------------------
// 6) (inst, class) histogram via LDS atomics, flushed to global
// ---------------------------------------------------------------------------
#define PIXPB (HWP / 128)
__global__ void __launch_bounds__(256)
k_counts(const int* __restrict__ segI, const int* __restrict__ closestI,
         unsigned* __restrict__ counts) {
  __shared__ unsigned lc[NBINS];
  int b = blockIdx.y;
  for (int i = threadIdx.x; i < NBINS; i += blockDim.x) lc[i] = 0u;
  __syncthreads();
  int pixStart = blockIdx.x * PIXPB;
  for (int i = threadIdx.x; i < PIXPB; i += blockDim.x) {
    size_t p = (size_t)b * HWP + pixStart + i;
    int s = segI[p];
    int cl = closestI[p];
    int inst = (s >= 24 && s <= 33) ? cl : 0;
    atomicAdd(&lc[inst * CC + s], 1u);
  }
  __syncthreads();
  unsigned* g = counts + (size_t)b * NBINS;
  for (int i = threadIdx.x; i < NBINS; i += blockDim.x) {
    unsigned v = lc[i];
    if (v) atomicAdd(&g[i], v);
  }
}

// ---------------------------------------------------------------------------
// 7) per-instance majority class, pixel count, validity
// ---------------------------------------------------------------------------
__global__ void __launch_bounds__(256)
k_inst(const unsigned* __restrict__ counts, const int* __restrict__ validk,
       int* __restrict__ instClsI, float* __restrict__ out) {
  int b = blockIdx.x;
  for (int k = threadIdx.x; k < KK; k += blockDim.x) {
    const unsigned* row = counts + (size_t)b * NBINS + (size_t)(k + 1) * CC;
    unsigned n = 0, bc = 0;
    int bcls = 0;
    for (int c = 0; c < CC; ++c) {
      unsigned v = row[c];
      n += v;
      if (v > bc) { bc = v; bcls = c; } // first-max == jnp.argmax
    }
    int vld = (n > 0) && validk[b * KK + k];
    int o = b * KK + k;
    instClsI[o] = bcls;
    out[O_CLASS + o] = (float)bcls;
    out[O_NPIX + o] = (float)n;
    out[O_VALID + o] = vld ? 1.f : 0.f;
  }
}

// ---------------------------------------------------------------------------
// 8) per-instance softmax-prob sums of the majority class (LDS partials)
// ---------------------------------------------------------------------------
__global__ void __launch_bounds__(256)
k_segprob(const float* __restrict__ logits, const float* __restrict__ maxlog,
          const float* __restrict__ invden, const int* __restrict__ segI,
          const int* __restrict__ closestI, const int* __restrict__ instClsI,
          float* __restrict__ sums) {
  __shared__ float ls[KK];
  int b = blockIdx.y;
  for (int i = threadIdx.x; i < KK; i += blockDim.x) ls[i] = 0.f;
  __syncthreads();
  int pixStart = blockIdx.x * PIXPB;
  for (int i = threadIdx.x; i < PIXPB; i += blockDim.x) {
    int pix = pixStart + i;
    size_t p = (size_t)b * HWP + pix;
    int s = segI[p];
    if (s >= 24 && s <= 33) {
      int inst = closestI[p]; // 1..K
      int c = instClsI[b * KK + inst - 1];
      float lg = logits[(size_t)b * CC * HWP + (size_t)c * HWP + pix];
      float prob = expf(lg - maxlog[p]) * invden[p];
      atomicAdd(&ls[inst - 1], prob);
    }
  }
  __syncthreads();
  for (int i = threadIdx.x; i < KK; i += blockDim.x) {
    float v = ls[i];
    if (v != 0.f) atomicAdd(&sums[b * KK + i], v);
  }
}

// ---------------------------------------------------------------------------
// 9) finalize mean seg prob
// ---------------------------------------------------------------------------
__global__ void k_fin(const float* __restrict__ sums, float* __restrict__ out) {
  int i = blockIdx.x * blockDim.x + threadIdx.x;
  if (i < BN * KK) {
    float n = out[O_NPIX + i];
    out[O_SPROB + i] = (n > 0.f) ? sums[i] / fmaxf(n, 1.f) : 0.f;
  }
}

// ---------------------------------------------------------------------------
extern "C" void kernel_launch(void* const* d_in, const int* in_sizes, int n_in,
                              void* d_out, int out_size, void* d_ws, size_t ws_size,
                              hipStream_t stream) {
  const float* segLogits = (const float*)d_in[0]; // (B,C,H,W)
  const float* centerMap = (const float*)d_in[1]; // (B,1,H,W)
  const float* regMaps   = (const float*)d_in[2]; // (B,2,H,W)
  float* out = (float*)d_out;

  char* ws = (char*)d_ws;
  size_t off = 0;
  auto take = [&](size_t bytes) -> void* {
    void* p = ws + off;
    off = (off + bytes + 255) & ~(size_t)255;
    return p;
  };
  float*    maxlog   = (float*)take((size_t)BN * HWP * 4);
  float*    invden   = (float*)take((size_t)BN * HWP * 4);
  int*      segI     = (int*)take((size_t)BN * HWP * 4);
  int*      closestI = (int*)take((size_t)BN * HWP * 4);
  float*    nms      = (float*)take((size_t)BN * HWP * 4);
  int*      candCnt  = (int*)take((size_t)BN * 4);
  float*    candVal  = (float*)take((size_t)BN * CAP * 4);
  int*      candIdx  = (int*)take((size_t)BN * CAP * 4);
  float*    probs    = (float*)take((size_t)BN * KK * 4);
  int*      validk   = (int*)take((size_t)BN * KK * 4);
  float*    ctab     = (float*)take((size_t)BN * CTE * 4);
  unsigned* counts   = (unsigned*)take((size_t)BN * NBINS * 4);
  float*    sums     = (float*)take((size_t)BN * KK * 4);
  int*      instClsI = (int*)take((size_t)BN * KK * 4);

  k_init<<<dim3((BN * NBINS + 255) / 256), dim3(256), 0, stream>>>(candCnt, counts, sums);
  k_softmax<<<dim3(HWP / 256, BN), dim3(256), 0, stream>>>(segLogits, out, maxlog, invden, segI);
  k_nms<<<dim3(WW / 16, HH / 16, BN), dim3(16, 16), 0, stream>>>(centerMap, nms);
  k_compact<<<dim3(HWP / 256, BN), dim3(256), 0, stream>>>(nms, candCnt, candVal, candIdx);
  k_topk<<<dim3(BN), dim3(256), 0, stream>>>(candCnt, candVal, candIdx,
                                             probs, validk, ctab, out);
  k_closest<<<dim3(HWP / (8 * 16), BN), dim3(256), 0, stream>>>(regMaps, ctab, closestI, out);
  k_counts<<<dim3(128, BN), dim3(256), 0, stream>>>(segI, closestI, counts);
  k_inst<<<dim3(BN), dim3(256), 0, stream>>>(counts, validk, instClsI, out);
  k_segprob<<<dim3(128, BN), dim3(256), 0, stream>>>(segLogits, maxlog, invden,
                                                     segI, closestI, instClsI, sums);
  k_fin<<<dim3((BN * KK + 255) / 256), dim3(256), 0, stream>>>(sums, out);
}